// Matching_14705968022077
// MI455X (gfx1250) — compile-verified
//
#include <hip/hip_runtime.h>
#include <stdint.h>

typedef __attribute__((ext_vector_type(2))) float v2f;
typedef __attribute__((ext_vector_type(8))) float v8f;

#define BATCH 16
#define CDIM  256
#define FDIM  4096
#define KC    32
#define KPAD  36   // +4 pad: keeps 16B row alignment, breaks bank aliasing

// ---------------- Kernel A: ynorm[b*C+m] = sum(y[b,m,:]^2) ----------------
__global__ __launch_bounds__(256)
void ynorm_kernel(const float* __restrict__ y, float* __restrict__ ynorm) {
    const int row = blockIdx.x;                      // 0 .. B*C-1
    const float* p = y + (size_t)row * FDIM;
    float s = 0.0f;
    for (int i = threadIdx.x; i < FDIM; i += 256) {
        float v = p[i];
        s += v * v;
    }
    #pragma unroll
    for (int m = 16; m >= 1; m >>= 1) s += __shfl_xor(s, m, 32);
    __shared__ float red[8];
    const int wave = threadIdx.x >> 5, lane = threadIdx.x & 31;
    if (lane == 0) red[wave] = s;
    __syncthreads();
    if (threadIdx.x == 0) {
        float t = 0.0f;
        #pragma unroll
        for (int w = 0; w < 8; ++w) t += red[w];
        ynorm[row] = t;
    }
}

// ------- Kernel B: per (batch, 16-row tile): GEMM + argmin + gather -------
__global__ __launch_bounds__(256)
void match_kernel(const float* __restrict__ x, const float* __restrict__ y,
                  const float* __restrict__ ynorm, float* __restrict__ out) {
    const int b       = blockIdx.x >> 4;     // C/16 = 16 row tiles per batch
    const int rowTile = blockIdx.x & 15;
    const int rowBase = rowTile * 16;

    __shared__ float As[2][16 * KPAD];
    __shared__ float Bs[2][CDIM * KPAD];
    __shared__ float redV[16][8];
    __shared__ int   redI[16][8];
    __shared__ int   finalIdx[16];

    const float* xb = x + ((size_t)b * CDIM + rowBase) * FDIM;
    const float* yb = y + (size_t)b * CDIM * FDIM;

    const int tid  = threadIdx.x;
    const int wave = tid >> 5;
    const int lane = tid & 31;
    const int half = lane >> 4;      // 0: lanes 0-15, 1: lanes 16-31
    const int l16  = lane & 15;

    // per-thread tile coordinates for the async staging copies
    const int ra  = tid >> 3;        // A tile: 16 rows x 8 float4  (tid < 128)
    const int ca4 = tid & 7;

    v8f acc0 = {};                   // candidates wave*32 + 0..15
    v8f acc1 = {};                   // candidates wave*32 + 16..31

    // Issue async global->LDS DMA for K-chunk starting at `kb` into buffer `buf`.
    auto issue_loads = [&](int kb, int buf) {
        if (tid < 128) {
            uint32_t lds = (uint32_t)(uintptr_t)&As[buf][ra * KPAD + ca4 * 4];
            uint32_t off = (uint32_t)((ra * FDIM + kb + ca4 * 4) * 4);
            asm volatile("global_load_async_to_lds_b128 %0, %1, %2"
                         :: "v"(lds), "v"(off), "s"(xb) : "memory");
        }
        #pragma unroll
        for (int i = 0; i < 8; ++i) {
            const int fid = tid + i * 256;     // 2048 float4 = 256 rows x 8
            const int r   = fid >> 3;
            const int c4  = fid & 7;
            uint32_t lds = (uint32_t)(uintptr_t)&Bs[buf][r * KPAD + c4 * 4];
            uint32_t off = (uint32_t)((r * FDIM + kb + c4 * 4) * 4);
            asm volatile("global_load_async_to_lds_b128 %0, %1, %2"
                         :: "v"(lds), "v"(off), "s"(yb) : "memory");
        }
    };

    // prologue: fill buffer 0
    issue_loads(0, 0);
    asm volatile("s_wait_asynccnt 0x0" ::: "memory");
    __syncthreads();

    int buf = 0;
    for (int kb = 0; kb < FDIM; kb += KC) {
        // prefetch next chunk into the other buffer (overlaps with WMMA below)
        if (kb + KC < FDIM) issue_loads(kb + KC, buf ^ 1);

        const int n0 = wave * 32;
        #pragma unroll
        for (int kk = 0; kk < KC; kk += 4) {
            const int kcol = kk + 2 * half;   // A lane layout: K = 2*(l/16)+{0,1}
            v2f a  = *(const v2f*)(&As[buf][l16 * KPAD + kcol]);
            v2f b0 = *(const v2f*)(&Bs[buf][(n0 + l16) * KPAD + kcol]);
            v2f b1 = *(const v2f*)(&Bs[buf][(n0 + 16 + l16) * KPAD + kcol]);
            acc0 = __builtin_amdgcn_wmma_f32_16x16x4_f32(
                       false, a, false, b0, (short)0, acc0, false, false);
            acc1 = __builtin_amdgcn_wmma_f32_16x16x4_f32(
                       false, a, false, b1, (short)0, acc1, false, false);
        }

        asm volatile("s_wait_asynccnt 0x0" ::: "memory");
        __syncthreads();
        buf ^= 1;
    }

    // ---- epilogue: score = ynorm[n] - 2*S ; min+argmin over candidates ----
    const int nA = wave * 32 + l16;
    const int nB = nA + 16;
    const float yn0 = ynorm[b * CDIM + nA];
    const float yn1 = ynorm[b * CDIM + nB];

    float bestV[8];
    int   bestI[8];
    #pragma unroll
    for (int v = 0; v < 8; ++v) {
        float s0 = yn0 - 2.0f * acc0[v];
        float s1 = yn1 - 2.0f * acc1[v];
        float val; int idx;
        if (s1 < s0) { val = s1; idx = nB; } else { val = s0; idx = nA; }
        // butterfly min within each 16-lane half (rows differ between halves)
        #pragma unroll
        for (int m = 1; m <= 8; m <<= 1) {
            float ov = __shfl_xor(val, m, 32);
            int   oi = __shfl_xor(idx, m, 32);
            if (ov < val || (ov == val && oi < idx)) { val = ov; idx = oi; }
        }
        bestV[v] = val; bestI[v] = idx;
    }

    if (l16 == 0) {                      // lanes 0 and 16 hold half results
        #pragma unroll
        for (int v = 0; v < 8; ++v) {
            const int m = v + 8 * half;  // local row
            redV[m][wave] = bestV[v];
            redI[m][wave] = bestI[v];
        }
    }
    __syncthreads();

    if (tid < 16) {                      // reduce across the 8 waves per row
        float bv = redV[tid][0]; int bi = redI[tid][0];
        #pragma unroll
        for (int w = 1; w < 8; ++w) {
            float ov = redV[tid][w]; int oi = redI[tid][w];
            if (ov < bv || (ov == bv && oi < bi)) { bv = ov; bi = oi; }
        }
        finalIdx[tid] = bi;
    }
    __syncthreads();

    // ---- gather: out[b, rowBase+r, :] = y[b, finalIdx[r], :] ----
    float* outb = out + ((size_t)b * CDIM + rowBase) * FDIM;
    for (int r = 0; r < 16; ++r) {
        const float* src = yb + (size_t)finalIdx[r] * FDIM;
        float*       dst = outb + (size_t)r * FDIM;
        for (int c4 = tid; c4 < FDIM / 4; c4 += 256) {
            *(float4*)(dst + c4 * 4) = *(const float4*)(src + c4 * 4);
        }
    }
}

extern "C" void kernel_launch(void* const* d_in, const int* in_sizes, int n_in,
                              void* d_out, int out_size, void* d_ws, size_t ws_size,
                              hipStream_t stream) {
    const float* x = (const float*)d_in[0];   // (B, C, H, W) f32
    const float* y = (const float*)d_in[1];   // (B, C, H, W) f32
    float* out   = (float*)d_out;             // (B, 256, H, W) f32
    float* ynorm = (float*)d_ws;              // B*C floats of scratch

    ynorm_kernel<<<BATCH * CDIM, 256, 0, stream>>>(y, ynorm);
    match_kernel<<<BATCH * (CDIM / 16), 256, 0, stream>>>(x, y, ynorm, out);
}